// FeatureSpaceRegularizationLoss_47682726920460
// MI455X (gfx1250) — compile-verified
//
#include <hip/hip_runtime.h>
#include <hip/hip_bf16.h>
#include <math.h>

typedef __attribute__((ext_vector_type(2))) float v2f;
typedef __attribute__((ext_vector_type(8))) float v8f;

#define NN 1024
#define TT 64
#define CC 256
#define NTILE 64          // 1024/16 tiles per matrix dimension
#define PROD_BLOCKS 256   // blocks in product pass

static __device__ __forceinline__ v8f wmma_f32_k4(v2f a, v2f b, v8f c) {
    // D = A(16x4,f32) * B(4x16,f32) + C(16x16,f32)
    return __builtin_amdgcn_wmma_f32_16x16x4_f32(
        /*neg_a=*/false, a, /*neg_b=*/false, b,
        /*c_mod=*/(short)0, c, /*reuse_a=*/false, /*reuse_b=*/false);
}

// Fast hardware sqrt: single v_sqrt_f32 (TRANS op, co-executes with VALU).
// Inputs are clamped to >= 1e-12, far from denormal, so no fixup needed.
static __device__ __forceinline__ float fast_sqrt(float x) {
    return __builtin_amdgcn_sqrtf(x);
}

// ---------------- Pass 1: mean over time (the only HBM-heavy pass) --------
__global__ __launch_bounds__(256) void k_mean(const float* __restrict__ o1,
                                              const float* __restrict__ ft,
                                              float* __restrict__ X,
                                              float* __restrict__ Y) {
    int idx = blockIdx.x * 256 + threadIdx.x;   // idx = n*CC + c
    int n = idx >> 8;
    int c = idx & 255;
    const float* p = o1 + (size_t)n * TT * CC + c;
    float s0 = 0.f;
    #pragma unroll 8
    for (int t = 0; t < TT; ++t) s0 += p[t * CC];
    X[idx] = s0 * (1.0f / TT);
    const float* q = ft + (size_t)n * TT * CC + c;
    float s1 = 0.f;
    #pragma unroll 8
    for (int t = 0; t < TT; ++t) s1 += q[t * CC];
    Y[idx] = s1 * (1.0f / TT);
}

// ---------------- Pass 2: row squared norms ------------------------------
__global__ __launch_bounds__(256) void k_norms(const float* __restrict__ X,
                                               const float* __restrict__ Y,
                                               float* __restrict__ nx,
                                               float* __restrict__ ny) {
    __shared__ float red[2][256];
    int n = blockIdx.x;
    int t = threadIdx.x;
    float vx = X[n * CC + t];
    float vy = Y[n * CC + t];
    red[0][t] = vx * vx;
    red[1][t] = vy * vy;
    __syncthreads();
    for (int s = 128; s > 0; s >>= 1) {
        if (t < s) {
            red[0][t] += red[0][t + s];
            red[1][t] += red[1][t + s];
        }
        __syncthreads();
    }
    if (t == 0) { nx[n] = red[0][0]; ny[n] = red[1][0]; }
}

// ---------------- Pass 3: WMMA Gram -> distance row sums -----------------
// Block b covers rows [b*16, b*16+16) x all 1024 cols. 8 waves; wave w does
// column tiles w, w+8, ..., w+56. Deterministic (no atomics).
__global__ __launch_bounds__(256) void k_rowsum(const float* __restrict__ Xm,
                                                const float* __restrict__ nrm,
                                                float* __restrict__ rs) {
    __shared__ float Apan[16 * CC];   // 16 KB A panel
    __shared__ float nxr[16];
    __shared__ float wpart[8][16];

    int i0 = blockIdx.x * 16;
    for (int t = threadIdx.x; t < 16 * CC; t += 256)
        Apan[t] = Xm[i0 * CC + t];
    if (threadIdx.x < 16) nxr[threadIdx.x] = nrm[i0 + threadIdx.x];
    __syncthreads();

    int wave = threadIdx.x >> 5;
    int lane = threadIdx.x & 31;
    int half = lane >> 4;       // 0: K/M-low group, 1: high group
    int m    = lane & 15;

    float rowacc[8];
    #pragma unroll
    for (int v = 0; v < 8; ++v) rowacc[v] = 0.f;

    for (int s = 0; s < 8; ++s) {
        int jt = wave + (s << 3);                       // column tile 0..63
        const float* Brow = Xm + (size_t)(jt * 16 + m) * CC;
        v8f acc = {};
        #pragma unroll 8
        for (int k = 0; k < CC; k += 4) {
            v2f a = *(const v2f*)(&Apan[m * CC + k + 2 * half]);
            v2f b = *(const v2f*)(Brow + k + 2 * half);
            acc = wmma_f32_k4(a, b, acc);
        }
        float nj = nrm[jt * 16 + m];
        #pragma unroll
        for (int v = 0; v < 8; ++v) {
            float ni = nxr[v + 8 * half];
            float ss = fmaxf(ni + nj - 2.0f * acc[v], 0.0f) + 1e-12f;
            rowacc[v] += fast_sqrt(ss);                 // d[i0+v+8*half][jt*16+m]
        }
    }
    // reduce each row across the 16-lane N-group
    #pragma unroll
    for (int v = 0; v < 8; ++v) {
        float r = rowacc[v];
        r += __shfl_xor(r, 1, 16);
        r += __shfl_xor(r, 2, 16);
        r += __shfl_xor(r, 4, 16);
        r += __shfl_xor(r, 8, 16);
        rowacc[v] = r;
    }
    if (m == 0) {
        #pragma unroll
        for (int v = 0; v < 8; ++v) wpart[wave][8 * half + v] = rowacc[v];
    }
    __syncthreads();
    if (threadIdx.x < 16) {
        float s = 0.f;
        #pragma unroll
        for (int w = 0; w < 8; ++w) s += wpart[w][threadIdx.x];
        rs[i0 + threadIdx.x] = s;
    }
}

// ---------------- Pass 4: grand sums -------------------------------------
__global__ __launch_bounds__(256) void k_grand(const float* __restrict__ rsx,
                                               const float* __restrict__ rsy,
                                               float* __restrict__ gs) {
    __shared__ float red[2][256];
    int t = threadIdx.x;
    float sx = 0.f, sy = 0.f;
    for (int i = t; i < NN; i += 256) { sx += rsx[i]; sy += rsy[i]; }
    red[0][t] = sx; red[1][t] = sy;
    __syncthreads();
    for (int s = 128; s > 0; s >>= 1) {
        if (t < s) { red[0][t] += red[0][t + s]; red[1][t] += red[1][t + s]; }
        __syncthreads();
    }
    if (t == 0) { gs[0] = red[0][0]; gs[1] = red[1][0]; }
}

// ---------------- Pass 5: centered products (both Grams per wave) --------
__global__ __launch_bounds__(256) void k_prod(const float* __restrict__ Xm,
                                              const float* __restrict__ Ym,
                                              const float* __restrict__ nx,
                                              const float* __restrict__ ny,
                                              const float* __restrict__ rsx,
                                              const float* __restrict__ rsy,
                                              const float* __restrict__ gs,
                                              float* __restrict__ partials) {
    int tid  = threadIdx.x;
    int wave = tid >> 5;
    int lane = tid & 31;
    int half = lane >> 4;
    int m    = lane & 15;

    const float invN  = 1.0f / (float)NN;
    const float invN2 = invN * invN;
    float gmx = gs[0] * invN2;
    float gmy = gs[1] * invN2;

    float sab = 0.f, saa = 0.f, sbb = 0.f;
    int waveg = blockIdx.x * 8 + wave;              // 0..2047, 2 tiles each

    for (int rep = 0; rep < 2; ++rep) {
        int tIdx = waveg * 2 + rep;                 // 0..4095
        int it = tIdx >> 6;
        int jt = tIdx & 63;
        const float* Ax = Xm + (size_t)(it * 16 + m) * CC;
        const float* Bx = Xm + (size_t)(jt * 16 + m) * CC;
        const float* Ay = Ym + (size_t)(it * 16 + m) * CC;
        const float* By = Ym + (size_t)(jt * 16 + m) * CC;
        v8f accx = {}, accy = {};
        #pragma unroll 4
        for (int k = 0; k < CC; k += 4) {
            int off = k + 2 * half;
            v2f ax = *(const v2f*)(Ax + off);
            v2f bx = *(const v2f*)(Bx + off);
            v2f ay = *(const v2f*)(Ay + off);
            v2f by = *(const v2f*)(By + off);
            accx = wmma_f32_k4(ax, bx, accx);
            accy = wmma_f32_k4(ay, by, accy);
        }
        float nxj = nx[jt * 16 + m], nyj = ny[jt * 16 + m];
        float rxj = rsx[jt * 16 + m] * invN;
        float ryj = rsy[jt * 16 + m] * invN;
        #pragma unroll
        for (int v = 0; v < 8; ++v) {
            int i = it * 16 + v + 8 * half;
            float dx = fast_sqrt(fmaxf(nx[i] + nxj - 2.0f * accx[v], 0.0f) + 1e-12f);
            float dy = fast_sqrt(fmaxf(ny[i] + nyj - 2.0f * accy[v], 0.0f) + 1e-12f);
            float a = dx - rsx[i] * invN - rxj + gmx;
            float b = dy - rsy[i] * invN - ryj + gmy;
            sab += a * b;
            saa += a * a;
            sbb += b * b;
        }
    }

    __shared__ float red[3][256];
    red[0][tid] = sab; red[1][tid] = saa; red[2][tid] = sbb;
    __syncthreads();
    for (int s = 128; s > 0; s >>= 1) {
        if (tid < s) {
            red[0][tid] += red[0][tid + s];
            red[1][tid] += red[1][tid + s];
            red[2][tid] += red[2][tid + s];
        }
        __syncthreads();
    }
    if (tid == 0) {
        partials[blockIdx.x * 3 + 0] = red[0][0];
        partials[blockIdx.x * 3 + 1] = red[1][0];
        partials[blockIdx.x * 3 + 2] = red[2][0];
    }
}

// ---------------- Pass 6: final scalar -----------------------------------
__global__ __launch_bounds__(256) void k_final(const float* __restrict__ partials,
                                               float* __restrict__ out) {
    __shared__ float red[3][256];
    int t = threadIdx.x;   // one partial-triple per thread (PROD_BLOCKS == 256)
    red[0][t] = partials[t * 3 + 0];
    red[1][t] = partials[t * 3 + 1];
    red[2][t] = partials[t * 3 + 2];
    __syncthreads();
    for (int s = 128; s > 0; s >>= 1) {
        if (t < s) {
            red[0][t] += red[0][t + s];
            red[1][t] += red[1][t + s];
            red[2][t] += red[2][t + s];
        }
        __syncthreads();
    }
    if (t == 0) {
        const float invN2 = 1.0f / ((float)NN * (float)NN);
        float xy = red[0][0] * invN2;
        float xx = red[1][0] * invN2;
        float yy = red[2][0] * invN2;
        float r = xy / sqrtf(xx * yy + 1e-9f);
        out[0] = 1.0f - r;
    }
}

extern "C" void kernel_launch(void* const* d_in, const int* in_sizes, int n_in,
                              void* d_out, int out_size, void* d_ws, size_t ws_size,
                              hipStream_t stream) {
    const float* o1 = (const float*)d_in[0];   // (1024, 64, 256) fp32
    const float* ft = (const float*)d_in[1];   // (1024, 64, 256) fp32
    float* out = (float*)d_out;                // scalar fp32

    float* ws = (float*)d_ws;                  // ~2.07 MB of fp32 scratch
    float* X        = ws;                      // 1024*256
    float* Y        = X + NN * CC;             // 1024*256
    float* nx       = Y + NN * CC;             // 1024
    float* ny       = nx + NN;                 // 1024
    float* rsx      = ny + NN;                 // 1024
    float* rsy      = rsx + NN;                // 1024
    float* gs       = rsy + NN;                // 2
    float* partials = gs + 2;                  // 256*3

    k_mean  <<<NN * CC / 256, 256, 0, stream>>>(o1, ft, X, Y);
    k_norms <<<NN,            256, 0, stream>>>(X, Y, nx, ny);
    k_rowsum<<<NTILE,         256, 0, stream>>>(X, nx, rsx);
    k_rowsum<<<NTILE,         256, 0, stream>>>(Y, ny, rsy);
    k_grand <<<1,             256, 0, stream>>>(rsx, rsy, gs);
    k_prod  <<<PROD_BLOCKS,   256, 0, stream>>>(X, Y, nx, ny, rsx, rsy, gs, partials);
    k_final <<<1,             256, 0, stream>>>(partials, out);
}